// GCNGraphClassifier_74758200754833
// MI455X (gfx1250) — compile-verified
//
#include <hip/hip_runtime.h>
#include <hip/hip_bf16.h>

#define GN_EPS 1e-5f

typedef __attribute__((ext_vector_type(2))) float v2f;
typedef __attribute__((ext_vector_type(8))) float v8f;

// ---------------- elementwise / init kernels ----------------

__global__ void fill_kernel(float* __restrict__ p, float v, int n) {
  int i = blockIdx.x * blockDim.x + threadIdx.x;
  if (i < n) p[i] = v;
}

// degree accumulation over real edges (self-loops pre-counted via fill(1.0))
__global__ void deg_kernel(const int* __restrict__ dst, float* __restrict__ deg, int E) {
  int i = blockIdx.x * blockDim.x + threadIdx.x;
  if (i < E) atomicAdd(&deg[dst[i]], 1.0f);
}

__global__ void rsqrt_kernel(float* __restrict__ p, int n) {
  int i = blockIdx.x * blockDim.x + threadIdx.x;
  if (i < n) p[i] = rsqrtf(p[i]);  // deg >= 1 always (self loops)
}

__global__ void gcnt_kernel(const int* __restrict__ batch, float* __restrict__ gcnt, int N) {
  int i = blockIdx.x * blockDim.x + threadIdx.x;
  if (i < N) atomicAdd(&gcnt[batch[i]], 1.0f);
}

// out[n,f] = b[f]
__global__ void init_bias_kernel(float* __restrict__ out, const float* __restrict__ b,
                                 int N, int F) {
  int i = blockIdx.x * blockDim.x + threadIdx.x;
  if (i < N * F) out[i] = b[i % F];
}

// ---------------- dense projections ----------------

// generic scalar GEMM (used for layer 1, K=3 — bandwidth trivial)
__global__ void gemm_scalar_kernel(const float* __restrict__ A, const float* __restrict__ W,
                                   float* __restrict__ C, int N, int K, int F) {
  int i = blockIdx.x * blockDim.x + threadIdx.x;
  if (i >= N * F) return;
  int n = i / F, f = i % F;
  float acc = 0.0f;
  for (int k = 0; k < K; ++k) acc += A[n * K + k] * W[k * F + f];
  C[i] = acc;
}

// fp32 WMMA GEMM: C[M,F] = A[M,K] @ W[K,F].  M%16==0, K%4==0, F%16==0.
// One wave owns one 16x16 output tile; 4 waves (128 thr) per block.
// Fragment layouts per CDNA5 ISA 7.12.2:
//   A 16x4 : lanes 0-15 -> M rows; VGPR0 holds K=0 (lo lanes)/K=2 (hi lanes), VGPR1 K=1/K=3
//   B 4x16 : VGPR r, lanes 0-15 -> K=r, N=lane; lanes 16-31 -> K=r+2
//   C 16x16: VGPR r, lanes 0-15 -> M=r, N=lane; lanes 16-31 -> M=r+8
__global__ void gemm_wmma_kernel(const float* __restrict__ A, const float* __restrict__ W,
                                 float* __restrict__ C, int M, int K, int F) {
  int wave = threadIdx.x >> 5;
  int lane = threadIdx.x & 31;
  int tileM = (blockIdx.x * 4 + wave) * 16;
  int tileN = blockIdx.y * 16;
  if (tileM >= M) return;                    // wave-uniform: EXEC stays all-1s below
  int half = lane >> 4;                      // 0 = lanes 0-15, 1 = lanes 16-31
  int l16  = lane & 15;

  v8f acc = {};
  const float* arow = A + (long long)(tileM + l16) * K;
  for (int k0 = 0; k0 < K; k0 += 4) {
    int ka = k0 + half * 2;                  // K index of fragment element 0 for this lane
    v2f a, b;
    a.x = arow[ka + 0];
    a.y = arow[ka + 1];
    b.x = W[(long long)(ka + 0) * F + tileN + l16];
    b.y = W[(long long)(ka + 1) * F + tileN + l16];
    acc = __builtin_amdgcn_wmma_f32_16x16x4_f32(
        /*neg_a=*/false, a, /*neg_b=*/false, b,
        /*c_mod=*/(short)0, acc, /*reuse_a=*/false, /*reuse_b=*/false);
  }
  float* crow = C + (long long)(tileM + half * 8) * F + tileN + l16;
#pragma unroll
  for (int r = 0; r < 8; ++r) crow[(long long)r * F] = acc[r];
}

// ---------------- message passing ----------------

// For each (edge, 4-feature chunk): out[dst] += dinv[src]*dinv[dst] * xw[src]
// Edges e < E come from edge_index; e >= E are self loops (src=dst=e-E).
__global__ void edge_scatter_kernel(const float* __restrict__ xw, const int* __restrict__ ei,
                                    const float* __restrict__ dinv, float* __restrict__ out,
                                    int E, int N, int F) {
  int chunks = F >> 2;
  long long i = (long long)blockIdx.x * blockDim.x + threadIdx.x;
  long long total = (long long)(E + N) * chunks;
  if (i >= total) return;
  int e = (int)(i / chunks);
  int f = ((int)(i % chunks)) << 2;
  int s, d;
  if (e < E) { s = ei[e]; d = ei[E + e]; } else { s = e - E; d = s; }
  float nrm = dinv[s] * dinv[d];
  const float4 v = *reinterpret_cast<const float4*>(xw + (long long)s * F + f);
  float* o = out + (long long)d * F + f;
  atomicAdd(o + 0, nrm * v.x);
  atomicAdd(o + 1, nrm * v.y);
  atomicAdd(o + 2, nrm * v.z);
  atomicAdd(o + 3, nrm * v.w);
}

// ---------------- GraphNorm ----------------

__global__ void seg_sum_kernel(const float* __restrict__ h, const int* __restrict__ batch,
                               float* __restrict__ gsum, int N, int F) {
  int i = blockIdx.x * blockDim.x + threadIdx.x;
  if (i >= N * F) return;
  int n = i / F, f = i % F;
  atomicAdd(&gsum[batch[n] * F + f], h[i]);
}

__global__ void seg_var_kernel(const float* __restrict__ h, const int* __restrict__ batch,
                               const float* __restrict__ gsum, const float* __restrict__ gcnt,
                               const float* __restrict__ alpha, float* __restrict__ gvar,
                               int N, int F) {
  int i = blockIdx.x * blockDim.x + threadIdx.x;
  if (i >= N * F) return;
  int n = i / F, f = i % F;
  int g = batch[n];
  float mean = gsum[g * F + f] / fmaxf(gcnt[g], 1.0f);
  float sub = h[i] - alpha[f] * mean;
  atomicAdd(&gvar[g * F + f], sub * sub);
}

__global__ void norm_relu_kernel(float* __restrict__ h, const int* __restrict__ batch,
                                 const float* __restrict__ gsum, const float* __restrict__ gvar,
                                 const float* __restrict__ gcnt, const float* __restrict__ alpha,
                                 const float* __restrict__ w, const float* __restrict__ b,
                                 int N, int F) {
  int i = blockIdx.x * blockDim.x + threadIdx.x;
  if (i >= N * F) return;
  int n = i / F, f = i % F;
  int g = batch[n];
  float cnt = fmaxf(gcnt[g], 1.0f);
  float mean = gsum[g * F + f] / cnt;
  float sub = h[i] - alpha[f] * mean;
  float istd = rsqrtf(gvar[g * F + f] / cnt + GN_EPS);
  h[i] = fmaxf(0.0f, w[f] * (sub * istd) + b[f]);
}

// ---------------- pooling + head ----------------

__global__ void pool_sum_kernel(const float* __restrict__ h, const int* __restrict__ batch,
                                float* __restrict__ pooled, int N, int F) {
  int i = blockIdx.x * blockDim.x + threadIdx.x;
  if (i >= N * F) return;
  int n = i / F, f = i % F;
  atomicAdd(&pooled[batch[n] * F + f], h[i]);
}

// out[g, j] = (pooled[g,:]/cnt[g]) . lin_w[:, j] + lin_b[j],  F=128, 3 outputs
__global__ void final_linear_kernel(const float* __restrict__ pooled, const float* __restrict__ gcnt,
                                    const float* __restrict__ lw, const float* __restrict__ lb,
                                    float* __restrict__ out, int G) {
  int i = blockIdx.x * blockDim.x + threadIdx.x;
  if (i >= G * 3) return;
  int g = i / 3, j = i % 3;
  float inv = 1.0f / fmaxf(gcnt[g], 1.0f);
  float acc = 0.0f;
  for (int f = 0; f < 128; ++f) acc += pooled[g * 128 + f] * inv * lw[f * 3 + j];
  out[i] = acc + lb[j];
}

// ---------------- host side ----------------

static inline int cdiv(long long a, int b) { return (int)((a + b - 1) / b); }

extern "C" void kernel_launch(void* const* d_in, const int* in_sizes, int n_in,
                              void* d_out, int out_size, void* d_ws, size_t ws_size,
                              hipStream_t stream) {
  const float* x      = (const float*)d_in[0];
  const int*   ei     = (const int*)d_in[1];   // [2, E] row-major
  const int*   batch  = (const int*)d_in[2];
  const float* W[3]   = {(const float*)d_in[3],  (const float*)d_in[8],  (const float*)d_in[13]};
  const float* bias[3]= {(const float*)d_in[4],  (const float*)d_in[9],  (const float*)d_in[14]};
  const float* gnw[3] = {(const float*)d_in[5],  (const float*)d_in[10], (const float*)d_in[15]};
  const float* gnb[3] = {(const float*)d_in[6],  (const float*)d_in[11], (const float*)d_in[16]};
  const float* gna[3] = {(const float*)d_in[7],  (const float*)d_in[12], (const float*)d_in[17]};
  const float* lin_w  = (const float*)d_in[18];
  const float* lin_b  = (const float*)d_in[19];
  float* out = (float*)d_out;

  const int N = in_sizes[2];
  const int E = in_sizes[1] / 2;
  const int G = 256;
  const int Fin[3]  = {3, 32, 64};
  const int Fout[3] = {32, 64, 128};

  // workspace carve-up (256B aligned)
  char* ws = (char*)d_ws;
  auto carve = [&](size_t bytes) {
    float* p = (float*)ws;
    ws += (bytes + 255) & ~(size_t)255;
    return p;
  };
  float* hbuf  = carve((size_t)N * 128 * sizeof(float));  // node features (in-place per layer)
  float* tmp   = carve((size_t)N * 128 * sizeof(float));  // xw scratch
  float* dinv  = carve((size_t)N * sizeof(float));
  float* gsum  = carve((size_t)G * 128 * sizeof(float));
  float* gvar  = carve((size_t)G * 128 * sizeof(float));
  float* gcnt  = carve((size_t)G * sizeof(float));
  float* pooled= carve((size_t)G * 128 * sizeof(float));

  const int B = 256;

  // degrees (self-loop contributes 1 to every node) -> dinv = deg^{-1/2}
  fill_kernel<<<cdiv(N, B), B, 0, stream>>>(dinv, 1.0f, N);
  deg_kernel<<<cdiv(E, B), B, 0, stream>>>(ei + E, dinv, E);
  rsqrt_kernel<<<cdiv(N, B), B, 0, stream>>>(dinv, N);

  // nodes per graph
  fill_kernel<<<cdiv(G, B), B, 0, stream>>>(gcnt, 0.0f, G);
  gcnt_kernel<<<cdiv(N, B), B, 0, stream>>>(batch, gcnt, N);

  const float* h_in = x;
  for (int l = 0; l < 3; ++l) {
    const int K = Fin[l], F = Fout[l];
    // 1) xw = h_in @ W  -> tmp
    if (l == 0) {
      gemm_scalar_kernel<<<cdiv((long long)N * F, B), B, 0, stream>>>(h_in, W[l], tmp, N, K, F);
    } else {
      dim3 grid(cdiv(N / 16, 4), F / 16);
      gemm_wmma_kernel<<<grid, 128, 0, stream>>>(h_in, W[l], tmp, N, K, F);
    }
    // 2) out = bias, then scatter-add normalized messages
    init_bias_kernel<<<cdiv((long long)N * F, B), B, 0, stream>>>(hbuf, bias[l], N, F);
    long long nscatter = (long long)(E + N) * (F >> 2);
    edge_scatter_kernel<<<cdiv(nscatter, B), B, 0, stream>>>(tmp, ei, dinv, hbuf, E, N, F);
    // 3) GraphNorm + ReLU (in place on hbuf)
    fill_kernel<<<cdiv(G * F, B), B, 0, stream>>>(gsum, 0.0f, G * F);
    seg_sum_kernel<<<cdiv((long long)N * F, B), B, 0, stream>>>(hbuf, batch, gsum, N, F);
    fill_kernel<<<cdiv(G * F, B), B, 0, stream>>>(gvar, 0.0f, G * F);
    seg_var_kernel<<<cdiv((long long)N * F, B), B, 0, stream>>>(hbuf, batch, gsum, gcnt, gna[l], gvar, N, F);
    norm_relu_kernel<<<cdiv((long long)N * F, B), B, 0, stream>>>(hbuf, batch, gsum, gvar, gcnt,
                                                                  gna[l], gnw[l], gnb[l], N, F);
    h_in = hbuf;
  }

  // global mean pool + classifier head
  fill_kernel<<<cdiv(G * 128, B), B, 0, stream>>>(pooled, 0.0f, G * 128);
  pool_sum_kernel<<<cdiv((long long)N * 128, B), B, 0, stream>>>(hbuf, batch, pooled, N, 128);
  final_linear_kernel<<<cdiv(G * 3, B), B, 0, stream>>>(pooled, gcnt, lin_w, lin_b, out, G);
}